// GCN_processor_29180007809052
// MI455X (gfx1250) — compile-verified
//
#include <hip/hip_runtime.h>

#define N_NODES 50000
#define N_EDGES 800000
#define DIM     128
#define LAYERS  3

typedef __attribute__((ext_vector_type(2))) float v2f;
typedef __attribute__((ext_vector_type(8))) float v8f;

// ---------------------------------------------------------------- utilities
__global__ void zero_kernel(float* __restrict__ p, int n) {
  int i = blockIdx.x * blockDim.x + threadIdx.x;
  if (i < n) p[i] = 0.0f;
}

// out_deg / in_deg accumulation: one thread per edge, fp32 hardware atomics.
__global__ void degree_kernel(const int* __restrict__ src,
                              const int* __restrict__ dst,
                              float* __restrict__ outdeg,
                              float* __restrict__ indeg, int e) {
  int i = blockIdx.x * blockDim.x + threadIdx.x;
  if (i < e) {
    __hip_atomic_fetch_add(&outdeg[src[i]], 1.0f, __ATOMIC_RELAXED,
                           __HIP_MEMORY_SCOPE_AGENT);
    __hip_atomic_fetch_add(&indeg[dst[i]], 1.0f, __ATOMIC_RELAXED,
                           __HIP_MEMORY_SCOPE_AGENT);
  }
}

// deg -> rsqrt(max(deg,1)) in place
__global__ void norm_kernel(float* __restrict__ d, int n) {
  int i = blockIdx.x * blockDim.x + threadIdx.x;
  if (i < n) d[i] = rsqrtf(fmaxf(d[i], 1.0f));
}

// ------------------------------------------------- edge gather + scatter-add
// wave32-per-edge: each lane owns 4 consecutive features (float4 load,
// 4 global_atomic_add_f32). h and agg are L2-resident (25.6 MB each vs 192 MB).
__global__ void __launch_bounds__(256)
gather_scatter_kernel(const float* __restrict__ h,
                      const int* __restrict__ src,
                      const int* __restrict__ dst,
                      const float* __restrict__ out_norm,
                      float* __restrict__ agg, int e) {
  int edge = blockIdx.x * 8 + (threadIdx.x >> 5);
  if (edge >= e) return;
  int lane = threadIdx.x & 31;
  int s = src[edge];
  int d = dst[edge];
  float scale = out_norm[s];
  const float4* hp = (const float4*)(h + (size_t)s * DIM) + lane;
  float4 v = *hp;
  float* ap = agg + (size_t)d * DIM + lane * 4;
  __hip_atomic_fetch_add(ap + 0, v.x * scale, __ATOMIC_RELAXED, __HIP_MEMORY_SCOPE_AGENT);
  __hip_atomic_fetch_add(ap + 1, v.y * scale, __ATOMIC_RELAXED, __HIP_MEMORY_SCOPE_AGENT);
  __hip_atomic_fetch_add(ap + 2, v.z * scale, __ATOMIC_RELAXED, __HIP_MEMORY_SCOPE_AGENT);
  __hip_atomic_fetch_add(ap + 3, v.w * scale, __ATOMIC_RELAXED, __HIP_MEMORY_SCOPE_AGENT);
}

// --------------------------------------------------- in_norm*(agg @ W) + b
// Identity used: (agg * in_norm[:,None]) @ W == in_norm[:,None] * (agg @ W),
// so the row scale moves to the 8-element accumulator epilogue and the LDS
// staging loop becomes a pure float4 copy (b128 load + b128 LDS store).
// One block = 16 rows x 128 cols; 8 waves, one 16x16 tile per wave via
// V_WMMA_F32_16X16X4_F32 over K=128 (32 steps). sA stride 132 floats keeps
// the strided fragment reads spread across LDS banks and rows 16B-aligned.
__global__ void __launch_bounds__(256)
gcn_matmul_wmma(const float* __restrict__ agg,
                const float* __restrict__ in_norm,
                const float* __restrict__ W,
                const float* __restrict__ bias,
                float* __restrict__ out, int relu) {
  __shared__ float sA[16 * 132];
  const int m0  = blockIdx.x * 16;
  const int tid = threadIdx.x;

  // stage A = agg[16 x 128] into LDS as float4 (512 vec4 / 256 threads)
  for (int j = tid; j < (16 * DIM) / 4; j += 256) {
    int row  = j >> 5;        // / 32 vec4 per row
    int col4 = j & 31;
    float4 v = ((const float4*)(agg + (size_t)(m0 + row) * DIM))[col4];
    *(float4*)&sA[row * 132 + col4 * 4] = v;
  }
  __syncthreads();

  const int wave  = tid >> 5;
  const int lane  = tid & 31;
  const int n0    = wave * 16;
  const int idx16 = lane & 15;          // A: row  /  B: column
  const int khalf = (lane >> 4) * 2;    // lanes 0-15 -> K{0,1}; 16-31 -> K{2,3}

  v8f acc = {};
  for (int k = 0; k < DIM; k += 4) {
    v2f a, b;
    a.x = sA[idx16 * 132 + k + khalf];
    a.y = sA[idx16 * 132 + k + khalf + 1];
    b.x = W[(size_t)(k + khalf) * DIM + n0 + idx16];
    b.y = W[(size_t)(k + khalf + 1) * DIM + n0 + idx16];
    // D = A(16x4) * B(4x16) + C   (fp32 WMMA, wave32)
    acc = __builtin_amdgcn_wmma_f32_16x16x4_f32(
        /*neg_a=*/false, a, /*neg_b=*/false, b,
        /*c_mod=*/(short)0, acc, /*reuse_a=*/false, /*reuse_b=*/false);
  }

  const float bv    = bias[n0 + idx16];
  const int   rbase = (lane >> 4) * 8;  // C/D layout: lanes 16-31 hold M+8
  for (int r = 0; r < 8; ++r) {
    int   row = m0 + rbase + r;
    float v   = fmaf(acc[r], in_norm[row], bv);
    if (relu) v = fmaxf(v, 0.0f);
    out[(size_t)row * DIM + n0 + idx16] = v;
  }
}

// --------------------------------------------------------------------- driver
extern "C" void kernel_launch(void* const* d_in, const int* in_sizes, int n_in,
                              void* d_out, int out_size, void* d_ws,
                              size_t ws_size, hipStream_t stream) {
  const float* h   = (const float*)d_in[0];   // [N,D]
  const float* e   = (const float*)d_in[1];   // [E,D] passthrough
  const int*   src = (const int*)d_in[2];     // [E]
  const int*   dst = (const int*)d_in[3];     // [E]
  const float* Ws  = (const float*)d_in[4];   // [L,D,D]
  const float* bs  = (const float*)d_in[5];   // [L,D]
  float*       out = (float*)d_out;           // h [N,D] then e [E,D]

  float* ws       = (float*)d_ws;
  float* out_norm = ws;                                   // N
  float* in_norm  = ws + N_NODES;                         // N
  float* hbuf     = ws + 2 * (size_t)N_NODES;             // N*D
  float* agg      = hbuf + (size_t)N_NODES * DIM;         // N*D

  const int ND = N_NODES * DIM;

  // degrees -> norms
  zero_kernel<<<(2 * N_NODES + 255) / 256, 256, 0, stream>>>(out_norm,
                                                             2 * N_NODES);
  degree_kernel<<<(N_EDGES + 255) / 256, 256, 0, stream>>>(src, dst, out_norm,
                                                           in_norm, N_EDGES);
  norm_kernel<<<(N_NODES + 255) / 256, 256, 0, stream>>>(out_norm, N_NODES);
  norm_kernel<<<(N_NODES + 255) / 256, 256, 0, stream>>>(in_norm, N_NODES);

  const float* hcur = h;
  for (int l = 0; l < LAYERS; ++l) {
    zero_kernel<<<(ND + 255) / 256, 256, 0, stream>>>(agg, ND);
    gather_scatter_kernel<<<(N_EDGES + 7) / 8, 256, 0, stream>>>(
        hcur, src, dst, out_norm, agg, N_EDGES);
    float* hout = (l == LAYERS - 1) ? out : hbuf;
    gcn_matmul_wmma<<<N_NODES / 16, 256, 0, stream>>>(
        agg, in_norm, Ws + (size_t)l * DIM * DIM, bs + (size_t)l * DIM, hout,
        (l < LAYERS - 1) ? 1 : 0);
    hcur = hbuf;
  }

  // e passthrough into the tail of d_out (dominant HBM traffic: ~820 MB)
  hipMemcpyAsync(out + (size_t)N_NODES * DIM, e,
                 (size_t)N_EDGES * DIM * sizeof(float),
                 hipMemcpyDeviceToDevice, stream);
}